// AggregateClusteredSum_2456721293662
// MI455X (gfx1250) — compile-verified
//
#include <hip/hip_runtime.h>

typedef __attribute__((ext_vector_type(2))) float v2f;
typedef __attribute__((ext_vector_type(8))) float v8f;

#define NBATCH   64
#define NTOK     4096
#define HDIM     256
#define NSEL     2048
#define KCLUST   32
#define ROWS_B   65                      // 2K+1
#define MTOT     (NBATCH * ROWS_B)       // 4160
#define GDIM     512

#define BS_STRIDE 288                    // interleaved B row stride (dwords), ==32 mod 64

// ---------------------------------------------------------------------------
// Kernel A: per-batch clustered segmented sum into LDS (ds_add_f32 atomics),
// then emit the 65 MLP input rows [Hk, Hk+hn, hn] and the cluster counts.
// ---------------------------------------------------------------------------
__global__ __launch_bounds__(256)
void build_x_kernel(const float* __restrict__ hs, const int* __restrict__ cs,
                    float* __restrict__ X, int* __restrict__ cnt_ws) {
  __shared__ float hk[KCLUST * HDIM];   // 32 KB cluster accumulators
  __shared__ int   csm[NSEL];           // 8 KB cluster ids
  __shared__ int   cnt[KCLUST];
  const int b    = blockIdx.x;
  const int t    = threadIdx.x;
  const int lane = t & 31;
  const int wave = t >> 5;

  for (int i = t; i < KCLUST * HDIM; i += 256) hk[i] = 0.0f;
  for (int i = t; i < NSEL; i += 256) csm[i] = cs[(size_t)b * NTOK + i];
  if (t < KCLUST) cnt[t] = 0;
  __syncthreads();

  // each wave owns 256 tokens; lane l handles columns [4l,4l+4) and [128+4l,...)
  const int tok0 = wave * (NSEL / 8);
  for (int i = tok0; i < tok0 + NSEL / 8; ++i) {
    const int k = csm[i];
    const float4* src = (const float4*)(hs + ((size_t)(b * NTOK + i)) * HDIM);
    const float4 v0 = src[lane];
    const float4 v1 = src[lane + 32];
    float* dst = &hk[k * HDIM + lane * 4];
    atomicAdd(&dst[0],   v0.x);
    atomicAdd(&dst[1],   v0.y);
    atomicAdd(&dst[2],   v0.z);
    atomicAdd(&dst[3],   v0.w);
    atomicAdd(&dst[128], v1.x);
    atomicAdd(&dst[129], v1.y);
    atomicAdd(&dst[130], v1.z);
    atomicAdd(&dst[131], v1.w);
    if (lane == 0) atomicAdd(&cnt[k], 1);
  }
  __syncthreads();

  // build the 65 rows for this batch (thread t = column)
  const float hn = hs[((size_t)(b * NTOK + NSEL)) * HDIM + t];
  float* xb = X + (size_t)b * ROWS_B * HDIM;
  for (int j = 0; j < KCLUST; ++j) {
    const float v = hk[j * HDIM + t];
    xb[(size_t)j * HDIM + t] = v;
    xb[(size_t)(KCLUST + j) * HDIM + t] = v + hn;
  }
  xb[(size_t)(2 * KCLUST) * HDIM + t] = hn;
  if (t < KCLUST) cnt_ws[b * KCLUST + t] = cnt[t];
}

// ---------------------------------------------------------------------------
// Kernel B: f32 WMMA GEMM, C = prelu(A[M x Kd] * W[Kd x N] + bias, alpha).
// 256 threads = 8 waves; block tile 64x128; wave tile 16x64 (4x v8f acc).
// K staged in LDS chunks of 32 with double buffering + register prefetch.
// B tile is k-pair interleaved so every B fragment is one ds_load_b64.
// ---------------------------------------------------------------------------
__global__ __launch_bounds__(256)
void gemm_prelu_wmma(const float* __restrict__ A, const float* __restrict__ W,
                     const float* __restrict__ bias, const float* __restrict__ alpha,
                     float* __restrict__ C, int N, int Kd, int act) {
  __shared__ float As[2][64][36];          // 2 x 9216 B
  __shared__ float Bs[2][16][BS_STRIDE];   // 2 x 18432 B, (k/2)-major, pair-interleaved
  const int t    = threadIdx.x;
  const int lane = t & 31;
  const int wave = t >> 5;
  const int l16  = lane & 15;
  const int hi   = lane >> 4;               // lane half (ISA frag layout)
  const int m0   = (wave & 3) * 16;
  const int n0   = (wave >> 2) * 64;
  const size_t bm = (size_t)blockIdx.x * 64;
  const int    bn = blockIdx.y * 128;

  v8f acc[4] = {};

  const int ar = t >> 3;            // 0..31  (A row within half-tile)
  const int ac = (t & 7) * 4;       // 0..28  (A col)
  const int wr = t >> 5;            // 0..7   (W row-pair, +8 on 2nd pass)
  const int wc = (t & 31) * 4;      // 0..124 (W col)

  // ---- load chunk 0 into LDS buffer 0 ----
  {
    const float4 a0 = *(const float4*)&A[(bm + ar) * Kd + ac];
    const float4 a1 = *(const float4*)&A[(bm + 32 + ar) * Kd + ac];
    *(float4*)&As[0][ar][ac]      = a0;
    *(float4*)&As[0][32 + ar][ac] = a1;
    #pragma unroll
    for (int p = 0; p < 2; ++p) {
      const int rp = wr + p * 8;    // row pair index 0..15 -> rows 2rp, 2rp+1
      const float4 w0 = *(const float4*)&W[(size_t)(2 * rp)     * N + bn + wc];
      const float4 w1 = *(const float4*)&W[(size_t)(2 * rp + 1) * N + bn + wc];
      float* dst = &Bs[0][rp][2 * wc];
      ((float4*)dst)[0] = make_float4(w0.x, w1.x, w0.y, w1.y);
      ((float4*)dst)[1] = make_float4(w0.z, w1.z, w0.w, w1.w);
    }
  }
  __syncthreads();

  int buf = 0;
  for (int kc = 0; kc < Kd; kc += 32) {
    const int has_next = (kc + 32 < Kd);
    // ---- prefetch next chunk into registers (latency overlapped with WMMA) ----
    float4 na0, na1, nw00, nw01, nw10, nw11;
    if (has_next) {
      const int kn = kc + 32;
      na0  = *(const float4*)&A[(bm + ar) * Kd + kn + ac];
      na1  = *(const float4*)&A[(bm + 32 + ar) * Kd + kn + ac];
      nw00 = *(const float4*)&W[(size_t)(kn + 2 * wr)      * N + bn + wc];
      nw01 = *(const float4*)&W[(size_t)(kn + 2 * wr + 1)  * N + bn + wc];
      nw10 = *(const float4*)&W[(size_t)(kn + 2 * (wr + 8))     * N + bn + wc];
      nw11 = *(const float4*)&W[(size_t)(kn + 2 * (wr + 8) + 1) * N + bn + wc];
    }
    // ---- compute current chunk ----
    #pragma unroll
    for (int k4 = 0; k4 < 8; ++k4) {
      // A 16x4 f32 frag: lanes 0-15 -> K = kb,kb+1 ; lanes 16-31 -> K+2,K+3
      const int kb = k4 * 4 + hi * 2;       // always even
      const v2f a = *(const v2f*)&As[buf][m0 + l16][kb];
      #pragma unroll
      for (int j = 0; j < 4; ++j) {
        // B 4x16 frag: (B[kb][n], B[kb+1][n]) contiguous in interleaved layout
        const v2f bf = *(const v2f*)&Bs[buf][kb >> 1][2 * (n0 + j * 16 + l16)];
        acc[j] = __builtin_amdgcn_wmma_f32_16x16x4_f32(
            false, a, false, bf, (short)0, acc[j], false, false);
      }
    }
    __syncthreads();
    // ---- store prefetched regs into the other buffer ----
    if (has_next) {
      const int nb = buf ^ 1;
      *(float4*)&As[nb][ar][ac]      = na0;
      *(float4*)&As[nb][32 + ar][ac] = na1;
      float* dst0 = &Bs[nb][wr][2 * wc];
      ((float4*)dst0)[0] = make_float4(nw00.x, nw01.x, nw00.y, nw01.y);
      ((float4*)dst0)[1] = make_float4(nw00.z, nw01.z, nw00.w, nw01.w);
      float* dst1 = &Bs[nb][wr + 8][2 * wc];
      ((float4*)dst1)[0] = make_float4(nw10.x, nw11.x, nw10.y, nw11.y);
      ((float4*)dst1)[1] = make_float4(nw10.z, nw11.z, nw10.w, nw11.w);
    }
    __syncthreads();
    buf ^= 1;
  }

  const float al = act ? alpha[0] : 0.0f;
  #pragma unroll
  for (int j = 0; j < 4; ++j) {
    const int n = bn + n0 + j * 16 + l16;
    const float bv = bias[n];
    #pragma unroll
    for (int v = 0; v < 8; ++v) {
      const size_t m = bm + m0 + hi * 8 + v;   // C/D: lanes 16-31 at M+8
      float val = acc[j][v] + bv;
      if (act) val = (val >= 0.0f) ? val : al * val;
      C[m * N + n] = val;
    }
  }
}

// ---------------------------------------------------------------------------
// Kernel C: combine gs -> (G, G_mask). Ks[b] = highest non-empty cluster.
// ---------------------------------------------------------------------------
__global__ __launch_bounds__(256)
void combine_kernel(const float* __restrict__ gs, const int* __restrict__ cnt_ws,
                    float* __restrict__ G, float* __restrict__ Gmask) {
  __shared__ float mk[KCLUST];
  __shared__ int ks_s;
  const int b = blockIdx.x;
  const int t = threadIdx.x;
  if (t < KCLUST) mk[t] = (cnt_ws[b * KCLUST + t] > 0) ? 1.0f : 0.0f;
  __syncthreads();
  if (t == 0) {
    int m = -1;
    for (int j = 0; j < KCLUST; ++j) if (mk[j] > 0.0f) m = j;
    ks_s = m;
  }
  __syncthreads();
  const int  Ksb  = ks_s;
  const bool need = (Ksb >= 0) && (Ksb < KCLUST - 1);

  for (int c = t; c < GDIM; c += 256) {
    const float* g = gs + (size_t)b * ROWS_B * GDIM + c;
    float S = 0.0f;
    for (int k = 0; k < KCLUST; ++k) S += mk[k] * g[(size_t)k * GDIM];
    const float glast = S + g[(size_t)(2 * KCLUST) * GDIM];
    for (int j = 0; j < KCLUST; ++j) {
      float val = mk[j] * (S - g[(size_t)j * GDIM] + g[(size_t)(KCLUST + j) * GDIM]);
      if (need && j == Ksb + 1) val = glast;   // "move" of G_last
      G[((size_t)b * (KCLUST + 1) + j) * GDIM + c] = val;
    }
    G[((size_t)b * (KCLUST + 1) + KCLUST) * GDIM + c] = need ? 0.0f : glast;
  }
  if (t < KCLUST + 1) {
    Gmask[b * (KCLUST + 1) + t] = (need && t >= Ksb + 2) ? 0.0f : 1.0f;
  }
}

// ---------------------------------------------------------------------------
extern "C" void kernel_launch(void* const* d_in, const int* in_sizes, int n_in,
                              void* d_out, int out_size, void* d_ws, size_t ws_size,
                              hipStream_t stream) {
  (void)in_sizes; (void)n_in; (void)out_size; (void)ws_size;
  const float* hs = (const float*)d_in[0];
  const float* W1 = (const float*)d_in[1];
  const float* b1 = (const float*)d_in[2];
  const float* a1 = (const float*)d_in[3];
  const float* W2 = (const float*)d_in[4];
  const float* b2 = (const float*)d_in[5];
  const float* a2 = (const float*)d_in[6];
  const float* W3 = (const float*)d_in[7];
  const float* b3 = (const float*)d_in[8];
  const float* a3 = (const float*)d_in[9];
  const float* W4 = (const float*)d_in[10];
  const float* b4 = (const float*)d_in[11];
  const float* a4 = (const float*)d_in[12];
  const float* W5 = (const float*)d_in[13];
  const float* b5 = (const float*)d_in[14];
  const float* a5 = (const float*)d_in[15];
  const float* W6 = (const float*)d_in[16];
  const float* b6 = (const float*)d_in[17];
  const int*   cs = (const int*)d_in[18];

  float* X0  = (float*)d_ws;                       // 4160 x 1024 ping
  float* X1  = X0 + (size_t)MTOT * 1024;           // 4160 x 1024 pong
  int*   cnt = (int*)(X1 + (size_t)MTOT * 1024);   // 64 x 32 counts

  float* G    = (float*)d_out;
  float* Gmsk = G + (size_t)NBATCH * (KCLUST + 1) * GDIM;

  build_x_kernel<<<NBATCH, 256, 0, stream>>>(hs, cs, X0, cnt);

  gemm_prelu_wmma<<<dim3(MTOT / 64, 1024 / 128), 256, 0, stream>>>(X0, W1, b1, a1, X1, 1024,  256, 1);
  gemm_prelu_wmma<<<dim3(MTOT / 64, 1024 / 128), 256, 0, stream>>>(X1, W2, b2, a2, X0, 1024, 1024, 1);
  gemm_prelu_wmma<<<dim3(MTOT / 64, 1024 / 128), 256, 0, stream>>>(X0, W3, b3, a3, X1, 1024, 1024, 1);
  gemm_prelu_wmma<<<dim3(MTOT / 64, 1024 / 128), 256, 0, stream>>>(X1, W4, b4, a4, X0, 1024, 1024, 1);
  gemm_prelu_wmma<<<dim3(MTOT / 64, 1024 / 128), 256, 0, stream>>>(X0, W5, b5, a5, X1, 1024, 1024, 1);
  gemm_prelu_wmma<<<dim3(MTOT / 64,  512 / 128), 256, 0, stream>>>(X1, W6, b6, nullptr, X0, 512, 1024, 0);

  combine_kernel<<<NBATCH, 256, 0, stream>>>(X0, cnt, G, Gmsk);
}